// MaskedBatchNorm2d_52733608461090
// MI455X (gfx1250) — compile-verified
//
#include <hip/hip_runtime.h>
#include <math.h>

typedef __attribute__((ext_vector_type(4))) float v4f;
typedef __attribute__((ext_vector_type(2))) float v2f;
typedef __attribute__((ext_vector_type(8))) float v8f;
typedef __attribute__((ext_vector_type(4))) int   v4i;

#define EPSV 1e-5f
#define B_N   16
#define C_CH  64
#define HW    65536   // 256*256

// --------------------------- CDNA5 async-to-LDS ----------------------------
#if defined(__has_builtin)
#if __has_builtin(__builtin_amdgcn_global_load_async_to_lds_b128)
#define HAVE_ASYNC_LDS 1
#endif
#endif
#ifndef HAVE_ASYNC_LDS
#define HAVE_ASYNC_LDS 0
#endif

#if HAVE_ASYNC_LDS
typedef __attribute__((address_space(1))) v4i GV4I;   // global int4
typedef __attribute__((address_space(3))) v4i LV4I;   // LDS int4
#if __has_builtin(__builtin_amdgcn_s_wait_asynccnt)
#define ASYNC_WAIT(n) __builtin_amdgcn_s_wait_asynccnt(n)
#else
#define ASYNC_WAIT(n) asm volatile("s_wait_asynccnt %0" ::"i"(n) : "memory")
#endif
#endif

// ---------------------------------------------------------------------------
// Kernel 1: per-(b,c) masked statistics. One 256-thread block per (b,c) slab
// (256 KB of x, contiguous). x is streamed via GLOBAL_LOAD_ASYNC_TO_LDS_B128
// (ASYNCcnt-tracked DMA, double-buffered per wave: each lane consumes exactly
// the 16B it fetched, so no workgroup barrier is required). The mask stays on
// the normal load path so its 4 MB footprint remains L2-resident across the
// 64 channel re-reads.
// ---------------------------------------------------------------------------
__global__ __launch_bounds__(256) void mbn_stats_kernel(
    const float* __restrict__ x, const float* __restrict__ mask,
    float* __restrict__ w_cnt, float* __restrict__ w_sum, float* __restrict__ w_sq) {
  const int bc  = blockIdx.x;        // 0..1023 = b*64 + c
  const int b   = bc >> 6;
  const int tid = threadIdx.x;
  const v4f* xv = (const v4f*)(x + (size_t)bc * HW);
  const v4f* mv = (const v4f*)(mask + (size_t)b * HW);

  float cnt = 0.f, s = 0.f, sq = 0.f;

#if HAVE_ASYNC_LDS
  __shared__ v4f stage[2][256];      // 2 x 4 KB double buffer
#define ISSUE(buf, it)                                                        \
  __builtin_amdgcn_global_load_async_to_lds_b128(                             \
      (GV4I*)(xv + (size_t)(it)*256 + tid), (LV4I*)&stage[buf][tid], 0, 0)

  ISSUE(0, 0);
  ISSUE(1, 1);
  for (int it = 0; it < 64; ++it) {
    if (it < 63) ASYNC_WAIT(1);      // oldest (it) landed, next may be in flight
    else         ASYNC_WAIT(0);      // last tile: drain fully
    __builtin_prefetch(&mv[(it + 2) * 256 + tid], 0, 0);
    v4f xx = stage[it & 1][tid];
    v4f mm = mv[it * 256 + tid];
#pragma unroll
    for (int j = 0; j < 4; ++j) {
      float m = mm[j] > 0.5f ? 1.f : 0.f;
      float v = xx[j];
      cnt += m;
      s  = fmaf(m, v, s);
      sq = fmaf(m * v, v, sq);
    }
    // Re-issue into the buffer just consumed. Safe: the accumulate above
    // forced s_wait_dscnt on the ds_load before this point.
    if (it + 2 < 64) ISSUE(it & 1, it + 2);
  }
#undef ISSUE
#else
  for (int i = tid; i < HW / 4; i += 256) {
    __builtin_prefetch(&xv[i + 256], 0, 0);          // global_prefetch_b8
    v4f xx = __builtin_nontemporal_load(&xv[i]);
    v4f mm = mv[i];
#pragma unroll
    for (int j = 0; j < 4; ++j) {
      float m = mm[j] > 0.5f ? 1.f : 0.f;
      float v = xx[j];
      cnt += m;
      s  = fmaf(m, v, s);
      sq = fmaf(m * v, v, sq);
    }
  }
#endif

  __shared__ float r0[256], r1[256], r2[256];
  r0[tid] = cnt; r1[tid] = s; r2[tid] = sq;
  __syncthreads();
  for (int off = 128; off > 0; off >>= 1) {
    if (tid < off) {
      r0[tid] += r0[tid + off];
      r1[tid] += r1[tid + off];
      r2[tid] += r2[tid + off];
    }
    __syncthreads();
  }
  if (tid == 0) { w_cnt[bc] = r0[0]; w_sum[bc] = r1[0]; w_sq[bc] = r2[0]; }
}

// ---------------------------------------------------------------------------
// Kernel 2: finalize. Compute mean_bc / var_bc (with the reference's cnt==0 /
// cnt<=1 edge cases), then reduce over the batch dimension (B=16) with
// V_WMMA_F32_16X16X4_F32: D = ones(16x4) x Bchunk(4x16) + C sums the K rows;
// 4 chained WMMAs cover all 16 batches; row M=0 (VGPR0, lanes 0-15) holds the
// per-channel sums. Single wave32 (EXEC all ones at every WMMA).
// ---------------------------------------------------------------------------
__global__ __launch_bounds__(32) void mbn_finalize_kernel(
    const float* __restrict__ w_cnt, const float* __restrict__ w_sum,
    const float* __restrict__ w_sq,  const float* __restrict__ weight,
    const float* __restrict__ bias,  float* __restrict__ w_scale,
    float* __restrict__ w_shift) {
  __shared__ float lmean[B_N * C_CH], lvar[B_N * C_CH];
  const int lane = threadIdx.x;

  for (int i = lane; i < B_N * C_CH; i += 32) {
    float cnt  = w_cnt[i], s = w_sum[i], sq = w_sq[i];
    float safe = fmaxf(cnt, 1.f);
    float mean = (cnt > 0.f) ? (s / safe) : 0.f;
    float d    = fmaxf(sq - mean * s, 0.f);   // = sum(m*(x-mean)^2)
    float var  = (cnt > 1.f) ? (d / safe) : 1.f;
    lmean[i] = mean;
    lvar[i]  = var;
  }
  __syncthreads();

  // B-matrix layout (4x16, K x N): lane = (K[1])*16 + N; VGPR0 <-> even K,
  // VGPR1 <-> odd K. A = all-ones, so its layout is immaterial.
  const int n  = lane & 15;
  const int k0 = (lane >> 4) << 1;   // 0 or 2
  v2f ones; ones.x = 1.f; ones.y = 1.f;

  for (int g = 0; g < 4; ++g) {      // 4 channel groups of 16
    v8f am = {}, av = {};
#pragma unroll
    for (int kb = 0; kb < 16; kb += 4) {   // batch chunks of K=4
      const int b0 = kb + k0, b1 = b0 + 1;
      const int col = g * 16 + n;
      v2f bm, bv;
      bm.x = lmean[b0 * C_CH + col]; bm.y = lmean[b1 * C_CH + col];
      bv.x = lvar [b0 * C_CH + col]; bv.y = lvar [b1 * C_CH + col];
      am = __builtin_amdgcn_wmma_f32_16x16x4_f32(false, ones, false, bm,
                                                 (short)0, am, false, false);
      av = __builtin_amdgcn_wmma_f32_16x16x4_f32(false, ones, false, bv,
                                                 (short)0, av, false, false);
    }
    if (lane < 16) {                 // D row M=0 -> VGPR0, lanes 0-15
      const int ch = g * 16 + lane;
      float fm = am[0] * (1.f / 16.f);
      float fv = av[0] * (1.f / 16.f);
      float sc = weight[ch] * rsqrtf(fv + EPSV);
      w_scale[ch] = sc;
      w_shift[ch] = fmaf(-fm, sc, bias[ch]);
    }
  }
}

// ---------------------------------------------------------------------------
// Kernel 3: normalize/apply. One float4 per thread; x and out are streamed
// non-temporally, mask read normally (reused 64x across channels -> L2 hits).
// ---------------------------------------------------------------------------
__global__ __launch_bounds__(256) void mbn_apply_kernel(
    const float* __restrict__ x, const float* __restrict__ mask,
    const float* __restrict__ w_scale, const float* __restrict__ w_shift,
    float* __restrict__ out) {
  const size_t gid = (size_t)blockIdx.x * 256 + threadIdx.x;  // float4 index
  const size_t i   = gid << 2;                                // element index
  const int    bc  = (int)(i >> 16);
  const int    ch  = bc & 63;
  const int    b   = bc >> 6;
  const size_t hw  = i & 65535;

  const v4f xx = __builtin_nontemporal_load((const v4f*)(x + i));
  const v4f mm = *(const v4f*)(mask + ((size_t)b << 16) + hw);
  const float sc = w_scale[ch];
  const float sh = w_shift[ch];

  v4f o;
#pragma unroll
  for (int j = 0; j < 4; ++j)
    o[j] = (mm[j] > 0.5f) ? fmaf(xx[j], sc, sh) : xx[j];
  __builtin_nontemporal_store(o, (v4f*)(out + i));
}

// ---------------------------------------------------------------------------
extern "C" void kernel_launch(void* const* d_in, const int* in_sizes, int n_in,
                              void* d_out, int out_size, void* d_ws, size_t ws_size,
                              hipStream_t stream) {
  (void)in_sizes; (void)n_in; (void)out_size; (void)ws_size;
  const float* x      = (const float*)d_in[0];
  const float* mask   = (const float*)d_in[1];
  const float* weight = (const float*)d_in[2];
  const float* bias   = (const float*)d_in[3];
  float* out = (float*)d_out;

  float* ws      = (float*)d_ws;
  float* w_cnt   = ws;                 // 1024
  float* w_sum   = ws + 1024;          // 1024
  float* w_sq    = ws + 2048;          // 1024
  float* w_scale = ws + 3072;          // 64
  float* w_shift = ws + 3136;          // 64

  mbn_stats_kernel<<<B_N * C_CH, 256, 0, stream>>>(x, mask, w_cnt, w_sum, w_sq);
  mbn_finalize_kernel<<<1, 32, 0, stream>>>(w_cnt, w_sum, w_sq, weight, bias,
                                            w_scale, w_shift);
  const int total_f4 = (B_N * C_CH * HW) / 4;     // 16,777,216
  mbn_apply_kernel<<<total_f4 / 256, 256, 0, stream>>>(x, mask, w_scale,
                                                       w_shift, out);
}